// FrameBlock_48326972015018
// MI455X (gfx1250) — compile-verified
//
#include <hip/hip_runtime.h>
#include <hip/hip_bf16.h>

// ---------------------------------------------------------------------------
// Graph-GRU on MI455X (gfx1250).  All node-propagation and 1x1-conv GEMMs go
// through V_WMMA_F32_16X16X4_F32 (fp32 in, fp32 accumulate — matches the
// fp32 reference).  One wave (32 lanes) per block computes a 16x64 C tile.
// Inner loop: branchless clamped-row A loads (b64), incremental B pointers.
// ---------------------------------------------------------------------------

typedef __attribute__((ext_vector_type(2))) float v2f;
typedef __attribute__((ext_vector_type(8))) float v8f;

#define TN 4          // 4 column tiles of 16 -> 64 columns per wave
#define NCOLS 1024    // N is always 1024 (node dimension)

__device__ __forceinline__ v8f wmma4(v2f a, v2f b, v8f c) {
  // D = A(16x4) * B(4x16) + C(16x16), fp32
  return __builtin_amdgcn_wmma_f32_16x16x4_f32(false, a, false, b,
                                               (short)0, c, false, false);
}

// Generic batched GEMM: C[z][r,n] = sum_k A[z][r,k] * B[z][k,n]  (N = 1024)
// epilogue: optional accumulate into existing C, bias[r], activation.
// act: 0=none, 1=prelu(*alpha_p), 2=sigmoid, 3=tanh
__global__ __launch_bounds__(32) void wmma_gemm_f32(
    const float* __restrict__ A, int lda, long long sA,
    const float* __restrict__ B, int ldb, long long sB,
    float* __restrict__ C, int ldc, long long sC,
    int M, int K,
    const float* __restrict__ bias, int act,
    const float* __restrict__ alpha_p, int accum, int vec2) {
  const int lane = threadIdx.x & 31;
  const int n0 = blockIdx.x * (16 * TN);
  const int m0 = blockIdx.y * 16;
  A += (long long)blockIdx.z * sA;
  B += (long long)blockIdx.z * sB;
  C += (long long)blockIdx.z * sC;

  // A fragment (16x4 fp32): lanes 0-15 hold K=0,1 ; lanes 16-31 hold K=2,3.
  // Rows >= M are address-clamped, not value-masked: garbage in A row m only
  // reaches D row m, and the epilogue never stores rows >= M.
  const int rowA = m0 + (lane & 15);
  const int rowSafe = (rowA < M) ? rowA : (M - 1);
  const int kb = (lane >> 4) << 1;  // 0 or 2
  const int colBase = n0 + (lane & 15);

  v8f acc[TN];
#pragma unroll
  for (int j = 0; j < TN; ++j) {
#pragma unroll
    for (int v = 0; v < 8; ++v) acc[j][v] = 0.0f;
  }

  const float* Ap = A + (long long)rowSafe * lda + kb;  // per-lane pair base
  const float* Bp = B + (long long)kb * ldb + colBase;  // per-lane column base
  const long long bStep = 4LL * ldb;

  int k0 = 0;
  if (vec2) {
    // lda % 4 == 0 and 16B-aligned base: each lane's (k,k+1) pair is one b64
    for (; k0 + 4 <= K; k0 += 4) {
      v2f a = *reinterpret_cast<const v2f*>(Ap);
#pragma unroll
      for (int j = 0; j < TN; ++j) {
        v2f b;
        b.x = Bp[j * 16];
        b.y = Bp[ldb + j * 16];
        acc[j] = wmma4(a, b, acc[j]);
      }
      Ap += 4;
      Bp += bStep;
    }
  } else {
    for (; k0 + 4 <= K; k0 += 4) {
      v2f a;
      a.x = Ap[0];
      a.y = Ap[1];
#pragma unroll
      for (int j = 0; j < TN; ++j) {
        v2f b;
        b.x = Bp[j * 16];
        b.y = Bp[ldb + j * 16];
        acc[j] = wmma4(a, b, acc[j]);
      }
      Ap += 4;
      Bp += bStep;
    }
  }
  // K remainder (K = 2, 65, 66 ...): out-of-range k must be value-zeroed
  // (it would otherwise pollute every output row).
  for (; k0 < K; k0 += 4) {
    const int kk = k0 + kb;
    v2f a;
    a.x = (kk < K) ? Ap[0] : 0.f;
    a.y = (kk + 1 < K) ? Ap[1] : 0.f;
#pragma unroll
    for (int j = 0; j < TN; ++j) {
      v2f b;
      b.x = (kk < K) ? Bp[j * 16] : 0.f;
      b.y = (kk + 1 < K) ? Bp[ldb + j * 16] : 0.f;
      acc[j] = wmma4(a, b, acc[j]);
    }
    Ap += 4;
    Bp += bStep;
  }

  // Epilogue.  C/D layout: lanes 0-15 -> rows m0+v, lanes 16-31 -> rows m0+8+v
  const float alpha = alpha_p ? *alpha_p : 0.f;
  const int rbase = m0 + ((lane >> 4) << 3);
#pragma unroll
  for (int v = 0; v < 8; ++v) {
    const int row = rbase + v;
    if (row >= M) continue;
    const long long off = (long long)row * ldc;
#pragma unroll
    for (int j = 0; j < TN; ++j) {
      const int col = colBase + j * 16;
      float val = acc[j][v];
      if (accum) val += C[off + col];
      if (bias) val += bias[row];
      if (act == 1) val = (val >= 0.f) ? val : alpha * val;
      else if (act == 2) val = 1.f / (1.f + __expf(-val));
      else if (act == 3) val = tanhf(val);
      C[off + col] = val;
    }
  }
}

// ---------------------------------------------------------------------------
// Elementwise / small kernels
// ---------------------------------------------------------------------------

// h[b,c,n] = h0[c,n]
__global__ void k_hinit(float* __restrict__ h, const float* __restrict__ h0) {
  int i = blockIdx.x * blockDim.x + threadIdx.x;  // 16*64*1024
  h[i] = h0[i & (64 * 1024 - 1)];
}

// Once per launch: W_cg = W_ci[:,1:] @ W_g ; fused biases ; coef2
__global__ void k_precomp(const float* __restrict__ W_ci, const float* __restrict__ W_g,
                          const float* __restrict__ b_ci, const float* __restrict__ b_g,
                          const float* __restrict__ b_ig, const float* __restrict__ W_ig,
                          float* __restrict__ W_cg, float* __restrict__ bias_comb,
                          float* __restrict__ coef2) {
  int o = blockIdx.x, i = threadIdx.x;  // 64 x 64
  float s = 0.f;
  for (int j = 0; j < 64; ++j) s += W_ci[o * 65 + 1 + j] * W_g[j * 64 + i];
  W_cg[o * 64 + i] = s;
  if (i == 0) {
    float bb = b_ci[o] + W_ci[o * 65] * b_ig[0];
    for (int j = 0; j < 64; ++j) bb += W_ci[o * 65 + 1 + j] * b_g[j];
    bias_comb[o] = bb;
    coef2[o] = W_ci[o * 65] * W_ig[0];
  }
}

// x_hat1 = W_ip . h ; preds out ; x_hat = mask ? x : x_hat1
__global__ void k_pred(const float* __restrict__ h, const float* __restrict__ x,
                       const unsigned char* __restrict__ mask,
                       const float* __restrict__ W_ip, const float* __restrict__ b_ip,
                       float* __restrict__ preds_out, float* __restrict__ xhat, int t) {
  int idx = blockIdx.x * blockDim.x + threadIdx.x;  // b*1024+n
  int b = idx >> 10, n = idx & 1023;
  const float* hb = h + (size_t)b * 65536 + n;
  float s = b_ip[0];
#pragma unroll
  for (int i = 0; i < 64; ++i) s += W_ip[i] * hb[i * 1024];
  size_t xo = (size_t)idx * 64 + t;
  preds_out[xo] = s;
  xhat[idx] = mask[xo] ? x[xo] : s;
}

// xr1[b,o,n] += coef2[o] * xp[b,n]    (folded ind_graph channel)
__global__ void k_rank1(float* __restrict__ xr1, const float* __restrict__ coef2,
                        const float* __restrict__ xp) {
  int idx = blockIdx.x * blockDim.x + threadIdx.x;  // 16*64*1024
  int b = idx >> 16, o = (idx >> 10) & 63, n = idx & 1023;
  xr1[idx] += coef2[o] * xp[(b << 10) + n];
}

// reprs[b,c,n,t] = c<64 ? xr2[b,c,n] : h[b,c-64,n]
__global__ void k_reprs(const float* __restrict__ xr2, const float* __restrict__ h,
                        float* __restrict__ reprs_out, int t) {
  int idx = blockIdx.x * blockDim.x + threadIdx.x;  // 16*128*1024
  int b = idx >> 17, c = (idx >> 10) & 127, n = idx & 1023;
  float v = (c < 64) ? xr2[(size_t)b * 65536 + c * 1024 + n]
                     : h[(size_t)b * 65536 + (c - 64) * 1024 + n];
  reprs_out[(size_t)idx * 64 + t] = v;
}

// x_hat2 = W_po . [xr2;h] ; imps out ; gin = [mask?x:x_hat2 , mask]
__global__ void k_postcore(const float* __restrict__ xr2, const float* __restrict__ h,
                           const float* __restrict__ x, const unsigned char* __restrict__ mask,
                           const float* __restrict__ W_po, const float* __restrict__ b_po,
                           float* __restrict__ imps_out, float* __restrict__ gin, int t) {
  int idx = blockIdx.x * blockDim.x + threadIdx.x;  // b*1024+n
  int b = idx >> 10, n = idx & 1023;
  const float* a = xr2 + (size_t)b * 65536 + n;
  const float* hb = h + (size_t)b * 65536 + n;
  float s = b_po[0];
#pragma unroll
  for (int o = 0; o < 64; ++o) s += W_po[o] * a[o * 1024] + W_po[64 + o] * hb[o * 1024];
  size_t xo = (size_t)idx * 64 + t;
  imps_out[xo] = s;
  unsigned char m = mask[xo];
  gin[(size_t)b * 2048 + n] = m ? x[xo] : s;
  gin[(size_t)b * 2048 + 1024 + n] = m ? 1.f : 0.f;
}

// rh = r * h
__global__ void k_rh(float* __restrict__ rh, const float* __restrict__ r,
                     const float* __restrict__ h) {
  int i = blockIdx.x * blockDim.x + threadIdx.x;
  rh[i] = r[i] * h[i];
}

// h_new = u*h + (1-u)*c
__global__ void k_hnew(float* __restrict__ hn, const float* __restrict__ u,
                       const float* __restrict__ h, const float* __restrict__ c) {
  int i = blockIdx.x * blockDim.x + threadIdx.x;
  float uu = u[i];
  hn[i] = uu * h[i] + (1.f - uu) * c[i];
}

// ---------------------------------------------------------------------------
extern "C" void kernel_launch(void* const* d_in, const int* in_sizes, int n_in,
                              void* d_out, int out_size, void* d_ws, size_t ws_size,
                              hipStream_t stream) {
  (void)in_sizes; (void)n_in; (void)out_size; (void)ws_size;
  const float* x       = (const float*)d_in[0];
  const unsigned char* mask = (const unsigned char*)d_in[1];
  const float* graph   = (const float*)d_in[2];
  const float* indg    = (const float*)d_in[3];
  const float* h0      = (const float*)d_in[4];
  const float* W_ip    = (const float*)d_in[5];
  const float* b_ip    = (const float*)d_in[6];
  const float* W_ig    = (const float*)d_in[7];
  const float* b_ig    = (const float*)d_in[8];
  const float* W_g     = (const float*)d_in[9];
  const float* b_g     = (const float*)d_in[10];
  const float* W_ci    = (const float*)d_in[11];
  const float* b_ci    = (const float*)d_in[12];
  const float* W_co    = (const float*)d_in[13];
  const float* b_co    = (const float*)d_in[14];
  const float* W_po    = (const float*)d_in[15];
  const float* b_po    = (const float*)d_in[16];
  const float* prelu_a = (const float*)d_in[17];
  const float* W_r     = (const float*)d_in[18];
  const float* b_r     = (const float*)d_in[19];
  const float* W_u     = (const float*)d_in[20];
  const float* b_u     = (const float*)d_in[21];
  const float* W_c     = (const float*)d_in[22];
  const float* b_c     = (const float*)d_in[23];

  float* out = (float*)d_out;
  float* out_imps  = out;                 // [16,1,1024,64]
  float* out_preds = out + 1048576;       // [16,1,1024,64]
  float* out_reprs = out + 2097152;       // [16,128,1024,64]

  // workspace carve (256B aligned)
  char* wsb = (char*)d_ws;
  size_t off = 0;
  auto alloc = [&](size_t nf) {
    float* p = (float*)(wsb + off);
    off += ((nf * 4 + 255) / 256) * 256;
    return p;
  };
  float* W_cg  = alloc(64 * 64);
  float* biasC = alloc(64);
  float* coef2 = alloc(64);
  float* hA    = alloc(16 * 64 * 1024);
  float* hB    = alloc(16 * 64 * 1024);
  float* xhat  = alloc(16 * 1024);
  float* xp    = alloc(16 * 1024);
  float* ghP   = alloc(16 * 66 * 1024);   // [ginP(2) ; h@graph(64)] per batch
  float* xr1   = alloc(16 * 64 * 1024);
  float* xr2   = alloc(16 * 64 * 1024);
  float* gin   = alloc(16 * 2 * 1024);
  float* rr    = alloc(16 * 64 * 1024);
  float* uu    = alloc(16 * 64 * 1024);
  float* rh    = alloc(16 * 64 * 1024);
  float* rhP   = alloc(16 * 64 * 1024);
  float* cc    = alloc(16 * 64 * 1024);

  auto gemm = [&](const float* A, int lda, long long sA,
                  const float* B, int ldb, long long sB,
                  float* C, int ldc, long long sC,
                  int M, int K, int batch,
                  const float* bias, int act, const float* alpha,
                  int accum, int vec2) {
    dim3 grid(NCOLS / (16 * TN), (M + 15) / 16, batch);
    wmma_gemm_f32<<<grid, dim3(32), 0, stream>>>(A, lda, sA, B, ldb, sB, C, ldc, sC,
                                                 M, K, bias, act, alpha, accum, vec2);
  };

  k_hinit<<<4096, 256, 0, stream>>>(hA, h0);
  k_precomp<<<64, 64, 0, stream>>>(W_ci, W_g, b_ci, b_g, b_ig, W_ig, W_cg, biasC, coef2);

  float* hc = hA;
  float* hn = hB;
  const long long SH = 65536, SG = 66 * 1024;

  for (int t = 0; t < 64; ++t) {
    // predict + first mask
    k_pred<<<64, 256, 0, stream>>>(hc, x, mask, W_ip, b_ip, out_preds, xhat, t);
    // xp = xhat @ ind_graph                       (M=16, K=1024)
    gemm(xhat, 1024, 0, indg, 1024, 0, xp, 1024, 0, 16, 1024, 1, nullptr, 0, nullptr, 0, 1);
    // ghP[2:66] = h @ graph                       (batched M=64, K=1024)
    gemm(hc, 1024, SH, graph, 1024, 0, ghP + 2048, 1024, SG, 64, 1024, 16,
         nullptr, 0, nullptr, 0, 1);
    // xr1 = W_cg @ (h@graph) + biasC              (M=64, K=64)
    gemm(W_cg, 64, 0, ghP + 2048, 1024, SG, xr1, 1024, SH, 64, 64, 16,
         biasC, 0, nullptr, 0, 1);
    // xr1 += coef2 (x) xp   (ind_graph channel)
    k_rank1<<<4096, 256, 0, stream>>>(xr1, coef2, xp);
    // xr2 = prelu(W_co[:,:64]@xr1 + W_co[:,64:]@h + b_co)
    gemm(W_co, 128, 0, xr1, 1024, SH, xr2, 1024, SH, 64, 64, 16, nullptr, 0, nullptr, 0, 1);
    gemm(W_co + 64, 128, 0, hc, 1024, SH, xr2, 1024, SH, 64, 64, 16,
         b_co, 1, prelu_a, 1, 1);
    // reprs output = [xr2 ; h]
    k_reprs<<<8192, 256, 0, stream>>>(xr2, hc, out_reprs, t);
    // x_hat2 (W_po), imps output, second mask, build gin
    k_postcore<<<64, 256, 0, stream>>>(xr2, hc, x, mask, W_po, b_po, out_imps, gin, t);
    // ghP[0:2] = gin @ graph                      (M=2, K=1024)
    gemm(gin, 1024, 2048, graph, 1024, 0, ghP, 1024, SG, 2, 1024, 16,
         nullptr, 0, nullptr, 0, 1);
    // r = sigmoid(W_r @ ghP + b_r)                (M=64, K=66)
    gemm(W_r, 66, 0, ghP, 1024, SG, rr, 1024, SH, 64, 66, 16, b_r, 2, nullptr, 0, 0);
    // u = sigmoid(W_u @ ghP + b_u)
    gemm(W_u, 66, 0, ghP, 1024, SG, uu, 1024, SH, 64, 66, 16, b_u, 2, nullptr, 0, 0);
    // rh = r * h ; rhP = rh @ graph
    k_rh<<<4096, 256, 0, stream>>>(rh, rr, hc);
    gemm(rh, 1024, SH, graph, 1024, 0, rhP, 1024, SH, 64, 1024, 16,
         nullptr, 0, nullptr, 0, 1);
    // c = tanh(W_c[:,:2]@ginP + W_c[:,2:]@rhP + b_c)
    gemm(W_c, 66, 0, ghP, 1024, SG, cc, 1024, SH, 64, 2, 16, nullptr, 0, nullptr, 0, 0);
    gemm(W_c + 2, 66, 0, rhP, 1024, SH, cc, 1024, SH, 64, 64, 16, b_c, 3, nullptr, 1, 0);
    // h update
    k_hnew<<<4096, 256, 0, stream>>>(hn, uu, hc, cc);
    float* tmp = hc; hc = hn; hn = tmp;
  }
}